// GraphSAGEBackbone_26731876451050
// MI455X (gfx1250) — compile-verified
//
#include <hip/hip_runtime.h>

#define NN 50000
#define NE 800000
#define FD 128

typedef __attribute__((ext_vector_type(2))) float v2f;
typedef __attribute__((ext_vector_type(8))) float v8f;

// ---------------- zero ----------------
__global__ __launch_bounds__(256) void k_zero(float* __restrict__ p, int n4) {
    int i = blockIdx.x * 256 + threadIdx.x;
    if (i < n4) ((float4*)p)[i] = make_float4(0.f, 0.f, 0.f, 0.f);
}

// ---------------- degree count ----------------
__global__ __launch_bounds__(256) void k_degree(const int* __restrict__ dst,
                                                float* __restrict__ cnt) {
    int e = blockIdx.x * 256 + threadIdx.x;
    if (e < NE) atomicAdd(&cnt[dst[e]], 1.0f);
}

// cnt -> 1/max(cnt,1)
__global__ __launch_bounds__(256) void k_rinv(float* __restrict__ cnt) {
    int i = blockIdx.x * 256 + threadIdx.x;
    if (i < NN) cnt[i] = 1.0f / fmaxf(cnt[i], 1.0f);
}

// ---------------- scatter-add: one wave per edge, float4 per lane ----------------
__global__ __launch_bounds__(256) void k_scatter(const float* __restrict__ X,
                                                 const int* __restrict__ src,
                                                 const int* __restrict__ dst,
                                                 float* __restrict__ S) {
    int t = blockIdx.x * 256 + threadIdx.x;
    int e = t >> 5;           // edge id
    int lane = t & 31;        // 32 lanes x float4 = 128 floats
    if (e >= NE) return;
    int s = src[e], d = dst[e];
    float4 v = ((const float4*)(X + (size_t)s * FD))[lane];
    float* o = S + (size_t)d * FD + lane * 4;
    atomicAdd(o + 0, v.x);
    atomicAdd(o + 1, v.y);
    atomicAdd(o + 2, v.z);
    atomicAdd(o + 3, v.w);
}

// ---------------- fused SAGE GEMM: out = mean@Wl^T + X@Wr^T + b, optional ReLU ----
// Block = 8 wave32s = 16 output rows x all 128 output cols.
// Wave w computes the 16x16 tile at columns [16w, 16w+16) using
// v_wmma_f32_16x16x4_f32 accumulation over K=128 (32 steps per matmul).
__global__ __launch_bounds__(256) void k_sage_gemm(
    const float* __restrict__ S, const float* __restrict__ rinv,
    const float* __restrict__ X,
    const float* __restrict__ Wl, const float* __restrict__ Wr,
    const float* __restrict__ bl, float* __restrict__ out, int relu) {
    __shared__ float sM[16][FD + 4];   // mean rows (pre-scaled), padded: conflict-free
    __shared__ float sX[16][FD + 4];   // raw input rows

    const int m0 = blockIdx.x * 16;
    const int tid = threadIdx.x;

    // Coalesced stage of 16 rows x 128 cols of S*rinv and X into LDS.
    for (int i = tid; i < 16 * FD; i += 256) {
        int r = i >> 7;            // /128
        int c = i & (FD - 1);
        int m = m0 + r;
        sM[r][c] = S[(size_t)m * FD + c] * rinv[m];
        sX[r][c] = X[(size_t)m * FD + c];
    }
    __syncthreads();

    const int wave = tid >> 5;
    const int lane = tid & 31;
    const int row  = lane & 15;            // A: M index / B: N index
    const int kg   = (lane >> 4) << 1;     // lane group K offset (0 or 2)
    const int ncol = wave * 16 + row;      // output column owned for B / bias / store

    const float* wl = Wl + (size_t)ncol * FD;  // row n of Wl == column n of Wl^T
    const float* wr = Wr + (size_t)ncol * FD;
    __builtin_prefetch(wl, 0, 0);              // global_prefetch_b8
    __builtin_prefetch(wr, 0, 0);

    // Seed accumulator with bias: every C/D element in this lane has N == ncol.
    const float bias = bl[ncol];
    v8f acc;
#pragma unroll
    for (int i = 0; i < 8; ++i) acc[i] = bias;

    // mean @ Wl^T
#pragma unroll
    for (int kb = 0; kb < FD; kb += 4) {
        v2f a = *(const v2f*)&sM[row][kb + kg];
        v2f b = *(const v2f*)&wl[kb + kg];
        acc = __builtin_amdgcn_wmma_f32_16x16x4_f32(
            /*neg_a=*/false, a, /*neg_b=*/false, b,
            /*c_mod=*/(short)0, acc, /*reuse_a=*/false, /*reuse_b=*/false);
    }
    // X @ Wr^T into the same accumulator
#pragma unroll
    for (int kb = 0; kb < FD; kb += 4) {
        v2f a = *(const v2f*)&sX[row][kb + kg];
        v2f b = *(const v2f*)&wr[kb + kg];
        acc = __builtin_amdgcn_wmma_f32_16x16x4_f32(
            false, a, false, b, (short)0, acc, false, false);
    }

    // C/D layout: VGPR v -> M = v + 8*(lane>=16), N = lane&15
    const int mbase = m0 + ((lane >> 4) << 3);
#pragma unroll
    for (int v = 0; v < 8; ++v) {
        float r = acc[v];
        if (relu) r = fmaxf(r, 0.f);
        out[(size_t)(mbase + v) * FD + ncol] = r;
    }
}

extern "C" void kernel_launch(void* const* d_in, const int* in_sizes, int n_in,
                              void* d_out, int out_size, void* d_ws, size_t ws_size,
                              hipStream_t stream) {
    const float* x   = (const float*)d_in[0];
    const int*   ei  = (const int*)d_in[1];   // edge_index [2, NE] (int32 under default JAX)
    const int*   src = ei;
    const int*   dst = ei + NE;
    const float* Wl0 = (const float*)d_in[2];
    const float* bl0 = (const float*)d_in[3];
    const float* Wr0 = (const float*)d_in[4];
    const float* Wl1 = (const float*)d_in[5];
    const float* bl1 = (const float*)d_in[6];
    const float* Wr1 = (const float*)d_in[7];
    const float* Wl2 = (const float*)d_in[8];
    const float* bl2 = (const float*)d_in[9];
    const float* Wr2 = (const float*)d_in[10];
    float* out = (float*)d_out;

    // Workspace layout (floats): S | rinv | hA | hB  (~77 MB total)
    float* ws   = (float*)d_ws;
    float* S    = ws;
    float* rinv = S + (size_t)NN * FD;          // 6,400,000 floats in
    float* hA   = rinv + 50048;                 // padded degree/rinv region
    float* hB   = hA + (size_t)NN * FD;

    dim3 b256(256);
    const int ZB4        = (NN * FD) / 4;       // 1,600,000 float4s
    const int ZBLK       = (ZB4 + 255) / 256;
    const int SC_BLOCKS  = (NE * 32) / 256;     // 100,000 (exact)
    const int GEMM_BLOCKS = NN / 16;            // 3,125 (exact)

    // Degrees -> reciprocal (shared by all three layers)
    k_zero  <<<(50048 / 4 + 255) / 256, b256, 0, stream>>>(rinv, 50048 / 4);
    k_degree<<<(NE + 255) / 256,        b256, 0, stream>>>(dst, rinv);
    k_rinv  <<<(NN + 255) / 256,        b256, 0, stream>>>(rinv);

    // Layer 0: x -> hA (ReLU)
    k_zero     <<<ZBLK,        b256, 0, stream>>>(S, ZB4);
    k_scatter  <<<SC_BLOCKS,   b256, 0, stream>>>(x, src, dst, S);
    k_sage_gemm<<<GEMM_BLOCKS, b256, 0, stream>>>(S, rinv, x, Wl0, Wr0, bl0, hA, 1);

    // Layer 1: hA -> hB (ReLU)
    k_zero     <<<ZBLK,        b256, 0, stream>>>(S, ZB4);
    k_scatter  <<<SC_BLOCKS,   b256, 0, stream>>>(hA, src, dst, S);
    k_sage_gemm<<<GEMM_BLOCKS, b256, 0, stream>>>(S, rinv, hA, Wl1, Wr1, bl1, hB, 1);

    // Layer 2: hB -> out (no ReLU)
    k_zero     <<<ZBLK,        b256, 0, stream>>>(S, ZB4);
    k_scatter  <<<SC_BLOCKS,   b256, 0, stream>>>(hB, src, dst, S);
    k_sage_gemm<<<GEMM_BLOCKS, b256, 0, stream>>>(S, rinv, hB, Wl2, Wr2, bl2, out, 0);
}